// NTN_56444460204420
// MI455X (gfx1250) — compile-verified
//
#include <hip/hip_runtime.h>

typedef __attribute__((ext_vector_type(2))) float v2f;
typedef __attribute__((ext_vector_type(8))) float v8f;

#define N_TOTAL 32768
#define DIM 128
#define K_SL 16
#define ROWS_PER_WAVE 16
#define WAVES_PER_BLOCK 8
#define ROWS_PER_BLOCK (ROWS_PER_WAVE * WAVES_PER_BLOCK)
#define EPS 1e-8f

// LDS: two matrices in WMMA-B fragment order.
//   frag[group][slot]  (float2 each)
//   group = et*32 + s   (et = 16-col output tile, s = 4-wide K-step)
//   slot  = h*16 + col  (h = lane>>4, col = lane&15)
//   slot content: {W[4s+2h][et*16+col], W[4s+2h+1][et*16+col]}
// -> inner-loop B load is ds_load_b64 at (base + (et*32+s)*256), conflict-free.

__global__ __launch_bounds__(256) void ntn_wmma_f32_kernel(
    const float* __restrict__ x1,
    const float* __restrict__ x2,
    const float* __restrict__ W1,
    const float* __restrict__ W2,
    const float* __restrict__ V,
    const float* __restrict__ b,
    float* __restrict__ out)
{
    extern __shared__ float smem[];          // 2 * 16384 floats = 128 KB

    const int k    = blockIdx.y;
    const int tid  = threadIdx.x;
    const int lane = tid & 31;
    const int wave = tid >> 5;
    const int half = lane >> 4;              // 0: lanes 0-15, 1: lanes 16-31
    const int mrow = lane & 15;

    // ---- Stage W1[k], W2[k] into LDS in fragment order (conflict-free b128 stores) ----
    {
        const int sub = lane & 15;
        const int h   = sub >> 3;            // which d-parity pair within the group
        const int c   = sub & 7;             // col pair index
        #pragma unroll 2
        for (int m = 0; m < 2; ++m) {
            const float* gW = (m ? W2 : W1) + (size_t)k * DIM * DIM;
            float* sF = smem + m * (DIM * DIM);
            for (int i = 0; i < 16; ++i) {
                const int grp = i * 16 + wave * 2 + half;   // 0..255, each exactly once
                const int et  = grp >> 5;
                const int s   = grp & 31;
                const int e   = et * 16 + 2 * c;
                const int d0  = 4 * s + 2 * h;
                v2f w0 = *(const v2f*)(gW + d0 * DIM + e);        // rows d0, d0+1
                v2f w1 = *(const v2f*)(gW + (d0 + 1) * DIM + e);
                float4 st = make_float4(w0.x, w1.x, w0.y, w1.y);  // slots 2c, 2c+1
                *(float4*)(sF + (size_t)(grp * 32 + h * 16 + 2 * c) * 2) = st;
            }
        }
    }
    __syncthreads();

    const int n_base = blockIdx.x * ROWS_PER_BLOCK + wave * ROWS_PER_WAVE;

    // ---- A fragments: 16x128 per side, ISA 32-bit 16x4 A layout per 4-wide K-step ----
    v2f a1[32], a2[32];
    {
        const float* r1 = x1 + (size_t)(n_base + mrow) * DIM;
        const float* r2 = x2 + (size_t)(n_base + mrow) * DIM;
        #pragma unroll
        for (int s = 0; s < 32; ++s) {
            const int d0 = 4 * s + 2 * half;
            a1[s] = *(const v2f*)(r1 + d0);
            a2[s] = *(const v2f*)(r2 + d0);
        }
    }

    v8f s12 = {}, s11 = {}, s22 = {};

    // ---- 8 e-tiles x 32 K-steps; B fragments are single ds_load_b64, immediate offsets ----
    const v2f* F1 = (const v2f*)smem + lane;             // side-1 fragment base
    const v2f* F2 = F1 + (DIM * DIM / 2);                // +8192 float2 = side 2
    for (int et = 0; et < 8; ++et) {
        const v2f* G1 = F1 + et * (32 * 32);
        const v2f* G2 = F2 + et * (32 * 32);
        v8f c1 = {}, c2 = {};
        #pragma unroll
        for (int s = 0; s < 32; ++s) {
            v2f b1 = G1[s * 32];                         // offset s*256 bytes (immediate)
            v2f b2 = G2[s * 32];
            c1 = __builtin_amdgcn_wmma_f32_16x16x4_f32(
                     false, a1[s], false, b1, (short)0, c1, false, false);
            c2 = __builtin_amdgcn_wmma_f32_16x16x4_f32(
                     false, a2[s], false, b2, (short)0, c2, false, false);
        }
        #pragma unroll
        for (int r = 0; r < 8; ++r) {
            s12[r] += c1[r] * c2[r];
            s11[r] += c1[r] * c1[r];
            s22[r] += c2[r] * c2[r];
        }
    }

    // ---- part2 = x_cat . V[k], from register-resident A fragments (v2f FMAs) ----
    v2f p2v = {};
    {
        const float* Vk = V + (size_t)k * 2 * DIM;
        #pragma unroll
        for (int s = 0; s < 32; ++s) {
            const int d0 = 4 * s + 2 * half;
            p2v += a1[s] * (*(const v2f*)(Vk + d0));
            p2v += a2[s] * (*(const v2f*)(Vk + DIM + d0));
        }
    }
    float p2 = p2v.x + p2v.y;
    p2 += __shfl_xor(p2, 16);        // lane l now holds full part2 for row (l & 15)

    // ---- finish D-reduction inside each 16-lane half (rows r in lanes 0-15, r+8 in 16-31) ----
    #pragma unroll
    for (int off = 8; off >= 1; off >>= 1) {
        #pragma unroll
        for (int r = 0; r < 8; ++r) {
            s12[r] += __shfl_xor(s12[r], off);
            s11[r] += __shfl_xor(s11[r], off);
            s22[r] += __shfl_xor(s22[r], off);
        }
    }

    // ---- epilogue: cosine + part2 + bias, ReLU, store out[n][k] ----
    const float bk = b[k];
    #pragma unroll
    for (int r = 0; r < 8; ++r) {
        const int row = r + 8 * half;
        const float n1 = fmaxf(sqrtf(s11[r]), EPS);
        const float n2 = fmaxf(sqrtf(s22[r]), EPS);
        const float cosv = s12[r] / (n1 * n2);
        const float p2r = __shfl(p2, row);   // lane 'row' holds part2 for that row
        float fea = cosv + p2r + bk;
        fea = fmaxf(fea, 0.0f);
        if (mrow == 0)
            out[(size_t)(n_base + row) * K_SL + k] = fea;
    }
}

extern "C" void kernel_launch(void* const* d_in, const int* in_sizes, int n_in,
                              void* d_out, int out_size, void* d_ws, size_t ws_size,
                              hipStream_t stream) {
    const float* x1 = (const float*)d_in[0];
    const float* x2 = (const float*)d_in[1];
    const float* W1 = (const float*)d_in[2];
    const float* W2 = (const float*)d_in[3];
    const float* V  = (const float*)d_in[4];
    const float* b  = (const float*)d_in[5];
    float* out = (float*)d_out;

    const size_t shmem = (size_t)2 * DIM * DIM * sizeof(float);   // 131072 B (< 320 KB WGP LDS)
    (void)hipFuncSetAttribute((const void*)ntn_wmma_f32_kernel,
                              hipFuncAttributeMaxDynamicSharedMemorySize,
                              (int)shmem);

    dim3 grid(N_TOTAL / ROWS_PER_BLOCK, K_SL);
    ntn_wmma_f32_kernel<<<grid, dim3(256), shmem, stream>>>(x1, x2, W1, W2, V, b, out);
}